// QCQPLayer_52312701665895
// MI455X (gfx1250) — compile-verified
//
#include <hip/hip_runtime.h>

#define BLK 256

typedef __attribute__((ext_vector_type(4))) float v4f;

__device__ __forceinline__ float frcp(float x)  { return __builtin_amdgcn_rcpf(x); }
__device__ __forceinline__ float frsq(float x)  { return __builtin_amdgcn_rsqf(x); }
__device__ __forceinline__ float fsqr(float x)  { return __builtin_amdgcn_sqrtf(x); }

// One Jacobi rotation on symmetric a[4][4] zeroing a[p][q]; accumulates V (columns = eigenvectors).
// NaN-safe: apq==0 (and the 0*inf path) collapses to t=0 -> identity rotation.
#define JROT(p, q)                                                         \
  do {                                                                     \
    float apq = a[p][q];                                                   \
    float app = a[p][p];                                                   \
    float aqq = a[q][q];                                                   \
    float th_ = (aqq - app) * 0.5f * frcp(apq);                            \
    float t_  = frcp(fabsf(th_) + fsqr(th_ * th_ + 1.0f));                 \
    t_ = (th_ < 0.0f) ? -t_ : t_;                                          \
    t_ = (t_ == t_) ? t_ : 0.0f; /* kill NaN from 0*inf */                 \
    float c_ = frsq(t_ * t_ + 1.0f);                                       \
    float s_ = t_ * c_;                                                    \
    a[p][p] = app - t_ * apq;                                              \
    a[q][q] = aqq + t_ * apq;                                              \
    a[p][q] = 0.0f;                                                        \
    a[q][p] = 0.0f;                                                        \
    _Pragma("unroll")                                                      \
    for (int k_ = 0; k_ < 4; ++k_) {                                       \
      if (k_ != p && k_ != q) {                                            \
        float akp = a[k_][p], akq = a[k_][q];                              \
        float nkp = c_ * akp - s_ * akq;                                   \
        float nkq = s_ * akp + c_ * akq;                                   \
        a[k_][p] = nkp; a[p][k_] = nkp;                                    \
        a[k_][q] = nkq; a[q][k_] = nkq;                                    \
      }                                                                    \
    }                                                                      \
    _Pragma("unroll")                                                      \
    for (int k_ = 0; k_ < 4; ++k_) {                                       \
      float vkp = v[k_][p], vkq = v[k_][q];                                \
      v[k_][p] = c_ * vkp - s_ * vkq;                                      \
      v[k_][q] = s_ * vkp + c_ * vkq;                                      \
    }                                                                      \
  } while (0)

// Compare-exchange eigenvalue i<->j (ascending) carrying eigenvector columns.
#define CSWAP(i, j)                                                        \
  do {                                                                     \
    if (d##i > d##j) {                                                     \
      float tt = d##i; d##i = d##j; d##j = tt;                             \
      _Pragma("unroll")                                                    \
      for (int k_ = 0; k_ < 4; ++k_) {                                     \
        float tv = v[k_][i]; v[k_][i] = v[k_][j]; v[k_][j] = tv;           \
      }                                                                    \
    }                                                                      \
  } while (0)

// Async global->LDS tile load: per CDNA5 ISA, INST_OFFSET is
// added to BOTH the LDS address in vdst and the global address, so one
// (laddr, goff) pair + literal offsets covers the whole 2560-float tile.
#define ASYNC_LD(OFS)                                                      \
  asm volatile("global_load_async_to_lds_b32 %0, %1, %2 offset:" #OFS      \
               :: "v"(laddr), "v"(goff), "s"(gbase) : "memory")

#define NTST4(x, y, z, w, ptr)                                             \
  __builtin_nontemporal_store((v4f){(x), (y), (z), (w)}, (v4f*)(ptr))

__global__ __launch_bounds__(BLK) void qcqp_eigh4_kernel(
    const float* __restrict__ theta, float* __restrict__ out, int n) {
  __shared__ float smem[BLK * 10];

  const unsigned tid = threadIdx.x;
  const size_t tile_elem0 = (size_t)blockIdx.x * (BLK * 10);  // flat float idx of tile start
  const size_t total = (size_t)n * 10;
  const int elem = (int)(blockIdx.x * BLK + tid);

  float t0 = 0.f, t1 = 0.f, t2 = 0.f, t3 = 0.f, t4 = 0.f;
  float t5 = 0.f, t6 = 0.f, t7 = 0.f, t8 = 0.f, t9 = 0.f;

  const bool full_tile = (tile_elem0 + (size_t)(BLK * 10) <= total);  // block-uniform
  if (full_tile) {
    // --- Fast path: stage 2560 floats coalesced into LDS with CDNA5 async
    // --- global->LDS loads (ASYNCcnt), zero per-lane clamp math.
    const float* gbase = theta + tile_elem0;
    const unsigned lds_base = (unsigned)(size_t)(void*)smem;  // low 32 bits of flat ptr are the LDS offset
    const unsigned goff  = tid * 4u;
    const unsigned laddr = lds_base + tid * 4u;
    ASYNC_LD(0);    ASYNC_LD(1024); ASYNC_LD(2048); ASYNC_LD(3072); ASYNC_LD(4096);
    ASYNC_LD(5120); ASYNC_LD(6144); ASYNC_LD(7168); ASYNC_LD(8192); ASYNC_LD(9216);
    asm volatile("s_wait_asynccnt 0x0" ::: "memory");
    __syncthreads();  // uniform: condition depends only on blockIdx

    t0 = smem[tid * 10 + 0]; t1 = smem[tid * 10 + 1];
    t2 = smem[tid * 10 + 2]; t3 = smem[tid * 10 + 3];
    t4 = smem[tid * 10 + 4]; t5 = smem[tid * 10 + 5];
    t6 = smem[tid * 10 + 6]; t7 = smem[tid * 10 + 7];
    t8 = smem[tid * 10 + 8]; t9 = smem[tid * 10 + 9];
  } else if (elem < n) {
    // --- Tail block (not taken for B = 1M): plain guarded loads.
    const float* p = theta + (size_t)elem * 10;
    t0 = p[0]; t1 = p[1]; t2 = p[2]; t3 = p[3]; t4 = p[4];
    t5 = p[5]; t6 = p[6]; t7 = p[7]; t8 = p[8]; t9 = p[9];
  }

  if (elem < n) {
    // --- Build symmetric A per _SYM_IDX.
    float a[4][4] = {{t0, t1, t2, t3},
                     {t1, t4, t5, t6},
                     {t2, t5, t7, t8},
                     {t3, t6, t8, t9}};
    float v[4][4] = {{1.f, 0.f, 0.f, 0.f},
                     {0.f, 1.f, 0.f, 0.f},
                     {0.f, 0.f, 1.f, 0.f},
                     {0.f, 0.f, 0.f, 1.f}};

    // --- 4 cyclic Jacobi sweeps (quadratic convergence; ample for fp32 on 4x4).
#pragma unroll
    for (int sweep = 0; sweep < 4; ++sweep) {
      JROT(0, 1); JROT(0, 2); JROT(0, 3);
      JROT(1, 2); JROT(1, 3); JROT(2, 3);
    }

    float d0 = a[0][0], d1 = a[1][1], d2 = a[2][2], d3 = a[3][3];
    // --- Sorting network, ascending (matches jnp.linalg.eigh ordering).
    CSWAP(0, 1); CSWAP(2, 3); CSWAP(0, 2); CSWAP(1, 3); CSWAP(1, 2);

    // --- q = eigenvector of smallest eigenvalue (column 0), deterministic sign.
    float q0 = v[0][0], q1 = v[1][0], q2 = v[2][0], q3 = v[3][0];
    float m = q0;
    m = (fabsf(q1) > fabsf(m)) ? q1 : m;
    m = (fabsf(q2) > fabsf(m)) ? q2 : m;
    m = (fabsf(q3) > fabsf(m)) ? q3 : m;
    float sg = (m < 0.0f) ? -1.0f : 1.0f;
    q0 *= sg; q1 *= sg; q2 *= sg; q3 *= sg;

    // --- Outputs: q (n*4) | A (n*16) | eigvals (n*4); 16B-aligned non-temporal
    // --- stores (write-once data, keep it out of the 192MB L2's way).
    float* qout = out;                       // q[elem] at elem*4 floats
    float* Aout = out + (size_t)n * 4;       // A[elem] at elem*16 floats
    float* eout = out + (size_t)n * 20;      // eig[elem] at elem*4 floats

    NTST4(q0, q1, q2, q3, qout + (size_t)elem * 4);
    NTST4(t0, t1, t2, t3, Aout + (size_t)elem * 16 + 0);
    NTST4(t1, t4, t5, t6, Aout + (size_t)elem * 16 + 4);
    NTST4(t2, t5, t7, t8, Aout + (size_t)elem * 16 + 8);
    NTST4(t3, t6, t8, t9, Aout + (size_t)elem * 16 + 12);
    NTST4(d0, d1, d2, d3, eout + (size_t)elem * 4);
  }
}

extern "C" void kernel_launch(void* const* d_in, const int* in_sizes, int n_in,
                              void* d_out, int out_size, void* d_ws, size_t ws_size,
                              hipStream_t stream) {
  (void)n_in; (void)out_size; (void)d_ws; (void)ws_size;
  const float* theta = (const float*)d_in[0];
  const int n = in_sizes[0] / 10;
  const int grid = (n + BLK - 1) / BLK;
  qcqp_eigh4_kernel<<<grid, BLK, 0, stream>>>(theta, (float*)d_out, n);
}